// SmallGCN_50723563766346
// MI455X (gfx1250) — compile-verified
//
#include <hip/hip_runtime.h>

typedef __attribute__((ext_vector_type(2))) float v2f;
typedef __attribute__((ext_vector_type(8))) float v8f;

#define HID 64

// ---------------------------------------------------------------------------
// degree / normalization
// ---------------------------------------------------------------------------
__global__ void deg_init_kernel(float* __restrict__ deg, int n) {
    int i = blockIdx.x * blockDim.x + threadIdx.x;
    if (i < n) deg[i] = 1.0f;   // self-loop contribution
}

__global__ void deg_count_kernel(const long long* __restrict__ dst,
                                 float* __restrict__ deg, int E) {
    int e = blockIdx.x * blockDim.x + threadIdx.x;
    if (e < E) atomicAdd(&deg[(int)dst[e]], 1.0f);
}

__global__ void deg_rsqrt_kernel(const float* __restrict__ deg,
                                 float* __restrict__ dis, int n) {
    int i = blockIdx.x * blockDim.x + threadIdx.x;
    if (i < n) dis[i] = __frsqrt_rn(deg[i]);   // deg >= 1 always
}

__global__ void zero_kernel(float* __restrict__ p, int n) {
    int i = blockIdx.x * blockDim.x + threadIdx.x;
    if (i < n) p[i] = 0.0f;
}

// ---------------------------------------------------------------------------
// WMMA GEMM:  D[M x 64] = A[M x K] * B[K x 64]   (fp32, V_WMMA_F32_16X16X4_F32)
// One wave per 16-row M tile, covering all four 16-col N tiles.
// Block = 160 threads = 5 waves.
// B staged in LDS pre-swizzled as [K/2][64][2] so each lane's K-pair fragment
// {B[k][n], B[k+1][n]} is one aligned ds_load_b64 (no VALU marshaling).
// ---------------------------------------------------------------------------
__global__ __launch_bounds__(160)
void gemm_wmma_kernel(const float* __restrict__ A, const float* __restrict__ B,
                      float* __restrict__ D, int M, int K) {
    __shared__ float Blds[128 * HID];

    const int tid = threadIdx.x;
    const int total = K * HID;
    for (int i = tid; i < total; i += 160) {
        const int p  = i >> 7;          // K-pair row
        const int r  = i & 127;
        const int n  = r >> 1;          // column 0..63
        const int kk = (p << 1) | (r & 1);
        Blds[i] = B[kk * HID + n];
    }
    __syncthreads();

    const int wave = tid >> 5;
    const int lane = tid & 31;
    const int tileM = blockIdx.x * 5 + wave;
    if (tileM * 16 >= M) return;                 // wave-uniform: EXEC stays full

    const int m      = tileM * 16 + (lane & 15);
    const int khalf  = (lane >> 4) << 1;         // 0 for lanes 0-15, 2 for 16-31
    const int ncol   = lane & 15;
    const float* __restrict__ arow = A + (size_t)m * K;

    v8f acc0 = {}, acc1 = {}, acc2 = {}, acc3 = {};

    for (int k = 0; k < K; k += 4) {
        v2f a;
        a.x = arow[k + khalf + 0];
        a.y = arow[k + khalf + 1];
        // swizzled B: pair-row (k+khalf)/2, column ncol (+16/+32/+48 per N tile)
        const v2f* __restrict__ bp =
            (const v2f*)&Blds[((k + khalf) >> 1) * 128 + 2 * ncol];

        acc0 = __builtin_amdgcn_wmma_f32_16x16x4_f32(false, a, false, bp[0],
                                                     (short)0, acc0, false, false);
        acc1 = __builtin_amdgcn_wmma_f32_16x16x4_f32(false, a, false, bp[16],
                                                     (short)0, acc1, false, false);
        acc2 = __builtin_amdgcn_wmma_f32_16x16x4_f32(false, a, false, bp[32],
                                                     (short)0, acc2, false, false);
        acc3 = __builtin_amdgcn_wmma_f32_16x16x4_f32(false, a, false, bp[48],
                                                     (short)0, acc3, false, false);
    }

    // C/D layout: VGPR v -> rows (v, v+8); lanes 0-15 / 16-31 select row halves
    const int rbase = tileM * 16 + ((lane >> 4) << 3);
#pragma unroll
    for (int v = 0; v < 8; ++v) {
        float* __restrict__ dr = D + (size_t)(rbase + v) * HID + ncol;
        dr[0]  = acc0[v];
        dr[16] = acc1[v];
        dr[32] = acc2[v];
        dr[48] = acc3[v];
    }
}

// ---------------------------------------------------------------------------
// Scatter-add:  out[dst] += h[src] * dis[src]*dis[dst]
// Work items = (E + N) edges (tail N = self-loops), 16 lanes/edge, 4 feats/lane
// ---------------------------------------------------------------------------
__global__ void scatter_kernel(const float* __restrict__ h,
                               const long long* __restrict__ src,
                               const long long* __restrict__ dst,
                               const float* __restrict__ dis,
                               float* __restrict__ out,
                               int E, int Nn) {
    const unsigned tid = blockIdx.x * blockDim.x + threadIdx.x;
    const unsigned e = tid >> 4;
    if (e >= (unsigned)(E + Nn)) return;
    const int fo = (int)(tid & 15u) << 2;

    int s, d;
    if (e < (unsigned)E) { s = (int)src[e]; d = (int)dst[e]; }
    else                 { s = d = (int)(e - (unsigned)E);   }

    const float w = dis[s] * dis[d];
    const float4 hv = *(const float4*)(h + (size_t)s * HID + fo);
    float* __restrict__ o = out + (size_t)d * HID + fo;
    atomicAdd(o + 0, hv.x * w);
    atomicAdd(o + 1, hv.y * w);
    atomicAdd(o + 2, hv.z * w);
    atomicAdd(o + 3, hv.w * w);
}

// ---------------------------------------------------------------------------
// In-place bias + ReLU over [N x 64]
// ---------------------------------------------------------------------------
__global__ void bias_relu_kernel(float* __restrict__ a,
                                 const float* __restrict__ b, int n) {
    int i = blockIdx.x * blockDim.x + threadIdx.x;
    if (i < n) {
        float v = a[i] + b[i & (HID - 1)];
        a[i] = v > 0.0f ? v : 0.0f;
    }
}

// ---------------------------------------------------------------------------
// Final linear layer: logits[N x 2] = z[N x 64] @ Wl[64 x 2] + bl
// ---------------------------------------------------------------------------
__global__ void final_linear_kernel(const float* __restrict__ z,
                                    const float* __restrict__ Wl,
                                    const float* __restrict__ bl,
                                    float* __restrict__ out, int n) {
    int i = blockIdx.x * blockDim.x + threadIdx.x;
    if (i >= n) return;
    const float* __restrict__ zr = z + (size_t)i * HID;
    float o0 = bl[0], o1 = bl[1];
#pragma unroll 4
    for (int f = 0; f < HID; ++f) {
        float v = zr[f];
        o0 = fmaf(v, Wl[2 * f + 0], o0);
        o1 = fmaf(v, Wl[2 * f + 1], o1);
    }
    out[2 * i + 0] = o0;
    out[2 * i + 1] = o1;
}

// ---------------------------------------------------------------------------
extern "C" void kernel_launch(void* const* d_in, const int* in_sizes, int n_in,
                              void* d_out, int out_size, void* d_ws, size_t ws_size,
                              hipStream_t stream) {
    const float*     x   = (const float*)d_in[0];
    const long long* ei  = (const long long*)d_in[1];
    const float*     W1  = (const float*)d_in[2];
    const float*     b1  = (const float*)d_in[3];
    const float*     W2  = (const float*)d_in[4];
    const float*     b2  = (const float*)d_in[5];
    const float*     Wl  = (const float*)d_in[6];
    const float*     bl  = (const float*)d_in[7];
    float*           out = (float*)d_out;

    const int Nn = in_sizes[0] / 128;   // 50000 nodes
    const int E  = in_sizes[1] / 2;     // 800000 edges
    const long long* src = ei;
    const long long* dst = ei + E;

    // workspace layout (all 16B-aligned: Nn*4 = 200000 bytes, multiple of 16)
    float* deg = (float*)d_ws;
    float* dis = deg + Nn;
    float* hb  = dis + Nn;                    // [Nn x 64] transform buffer
    float* ag  = hb + (size_t)Nn * HID;       // [Nn x 64] aggregation buffer

    const int T = 256;
    const int nfeat = Nn * HID;
    const int mTiles = (Nn + 15) / 16;                       // 3125
    const int gemmBlocks = (mTiles + 4) / 5;                 // 625
    const unsigned scatterThreads = (unsigned)(E + Nn) * 16u;

    // normalization
    deg_init_kernel <<<(Nn + T - 1) / T, T, 0, stream>>>(deg, Nn);
    deg_count_kernel<<<(E  + T - 1) / T, T, 0, stream>>>(dst, deg, E);
    deg_rsqrt_kernel<<<(Nn + T - 1) / T, T, 0, stream>>>(deg, dis, Nn);

    // layer 1: h = x @ W1 ; agg = scatter(h*norm) ; z1 = relu(agg + b1)
    gemm_wmma_kernel<<<gemmBlocks, 160, 0, stream>>>(x, W1, hb, Nn, 128);
    zero_kernel     <<<(nfeat + T - 1) / T, T, 0, stream>>>(ag, nfeat);
    scatter_kernel  <<<(scatterThreads + T - 1) / T, T, 0, stream>>>(
        hb, src, dst, dis, ag, E, Nn);
    bias_relu_kernel<<<(nfeat + T - 1) / T, T, 0, stream>>>(ag, b1, nfeat);

    // layer 2: h = z1 @ W2 ; agg = scatter(h*norm) ; z2 = relu(agg + b2)
    gemm_wmma_kernel<<<gemmBlocks, 160, 0, stream>>>(ag, W2, hb, Nn, 64);
    zero_kernel     <<<(nfeat + T - 1) / T, T, 0, stream>>>(ag, nfeat);
    scatter_kernel  <<<(scatterThreads + T - 1) / T, T, 0, stream>>>(
        hb, src, dst, dis, ag, E, Nn);
    bias_relu_kernel<<<(nfeat + T - 1) / T, T, 0, stream>>>(ag, b2, nfeat);

    // output layer
    final_linear_kernel<<<(Nn + T - 1) / T, T, 0, stream>>>(ag, Wl, bl, out, Nn);
}